// GroupBatchNorm_16836271800620
// MI455X (gfx1250) — compile-verified
//
#include <hip/hip_runtime.h>
#include <hip/hip_bf16.h>

typedef float v2f __attribute__((ext_vector_type(2)));
typedef float v4f __attribute__((ext_vector_type(4)));
typedef float v8f __attribute__((ext_vector_type(8)));

#define EPS_GBN 1e-5f

// ---------------------------------------------------------------------------
// Kernel 1: zero the group accumulators (s1, s2, cnt). Must run every launch
// (graph replay does not re-initialize workspace).
// ---------------------------------------------------------------------------
__global__ void gbn_zero(float* __restrict__ p, int n) {
    int i = blockIdx.x * blockDim.x + threadIdx.x;
    if (i < n) p[i] = 0.0f;
}

// ---------------------------------------------------------------------------
// Kernel 2: batch reduction via f32 WMMA + segment atomics.
// One wave (32 threads) per (16-channel tile x batch slice). A = ones(16x4),
// B = 4x16 tile of x.  D = A*B + C -> every row of D holds the 16 column sums
// (exact f32 math).  With A = all-ones the result is invariant to the
// K-packing of the B operand; only the C/D layout (N = lane 0..15, M=0 in
// VGPR0) is needed for extraction.  gridDim.y slices the batch for occupancy;
// each slice atomically adds its partial sums (and its own row count).
//
// The loop body is branch-free (unconditional speculative prefetch, running
// pointer instead of per-iteration index math) so the compiler can unroll 4x
// and keep 8 loads in flight with staged s_wait_loadcnt.
// ---------------------------------------------------------------------------
__global__ void __launch_bounds__(32)
gbn_colsum_wmma(const float* __restrict__ x,
                const int*   __restrict__ cg,
                float* __restrict__ s1,
                float* __restrict__ s2,
                float* __restrict__ cnt,
                int Bdim, int Cdim, int rowsPerSlice) {
    const int c0   = blockIdx.x << 4;     // 16 channels per wave
    const int lane = threadIdx.x;         // 0..31 (wave32)
    const int half = lane >> 4;           // which K-slice pair this lane feeds
    const int n    = lane & 15;           // column within tile

    int bBegin = blockIdx.y * rowsPerSlice;
    int bEnd   = bBegin + rowsPerSlice;
    if (bEnd > Bdim) bEnd = Bdim;

    v2f a;  a.x = 1.0f; a.y = 1.0f;       // A = ones(16x4): 2 VGPRs, all 1.0
    v8f acc1 = {};                        // accumulates sum(x)
    v8f acc2 = {};                        // accumulates sum(x*x)

    const size_t strideRow = (size_t)Cdim;        // floats per batch row
    const float* p = x + (size_t)c0 + (size_t)n
                       + (size_t)(bBegin + 2 * half) * strideRow;

#pragma unroll 4
    for (int b0 = bBegin; b0 < bEnd; b0 += 4) {
        // speculative prefetch 4 iterations ahead (global_prefetch_b8);
        // harmless past the end (translation failures silently dropped)
        __builtin_prefetch(p + 16 * strideRow, 0, 3);
        float e0 = p[0];
        float e1 = p[strideRow];
        v2f bv;  bv.x  = e0;      bv.y  = e1;
        v2f bv2; bv2.x = e0 * e0; bv2.y = e1 * e1;
        acc1 = __builtin_amdgcn_wmma_f32_16x16x4_f32(
            false, a, false, bv,  (short)0, acc1, false, false);
        acc2 = __builtin_amdgcn_wmma_f32_16x16x4_f32(
            false, a, false, bv2, (short)0, acc2, false, false);
        p += 4 * strideRow;
    }

    // C/D layout: VGPR0, lanes 0..15 = (M=0, N=lane). All rows are identical.
    if (lane < 16) {
        const int c = c0 + lane;
        const int g = cg[c];
        atomicAdd(&s1[g], acc1[0]);
        atomicAdd(&s2[g], acc2[0]);
        atomicAdd(&cnt[g], (float)(bEnd - bBegin));
    }
}

// ---------------------------------------------------------------------------
// Kernel 3: per-group affine params.  scale = gamma * rsqrt(var+eps),
// bias = beta - mean*scale  (so the hot pass is a single FMA per element).
// ---------------------------------------------------------------------------
__global__ void gbn_group_params(const float* __restrict__ s1,
                                 const float* __restrict__ s2,
                                 const float* __restrict__ cnt,
                                 const float* __restrict__ gamma,
                                 const float* __restrict__ beta,
                                 float* __restrict__ scaleG,
                                 float* __restrict__ biasG, int G) {
    int g = blockIdx.x * blockDim.x + threadIdx.x;
    if (g >= G) return;
    float nn   = fmaxf(cnt[g], 1.0f);
    float mean = s1[g] / nn;
    float var  = s2[g] / nn - mean * mean;
    float inv  = rsqrtf(var + EPS_GBN);
    float sc   = gamma[g] * inv;
    scaleG[g]  = sc;
    biasG[g]   = beta[g] - mean * sc;
}

// ---------------------------------------------------------------------------
// Kernel 3b: expand group params to per-channel arrays (256 KB each; they stay
// L2/WGP$-resident and get reused by all batch rows, removing the dependent
// cg->group gather from the bandwidth-bound pass).
// ---------------------------------------------------------------------------
__global__ void gbn_channel_params(const int* __restrict__ cg,
                                   const float* __restrict__ scaleG,
                                   const float* __restrict__ biasG,
                                   float* __restrict__ scaleC,
                                   float* __restrict__ biasC, int C) {
    int c = blockIdx.x * blockDim.x + threadIdx.x;
    if (c >= C) return;
    int g = cg[c];
    scaleC[c] = scaleG[g];
    biasC[c]  = biasG[g];
}

// ---------------------------------------------------------------------------
// Kernel 4: normalize, float4-vectorized, 2D grid (x = channel quads,
// y = batch row) so the channel index needs no modulo.  Output is streamed
// with non-temporal stores (TH=NT) so the 128 MiB write doesn't evict x from
// the 192 MB L2 — x re-reads then hit L2 at full rate.
// ---------------------------------------------------------------------------
__global__ void gbn_normalize(const float* __restrict__ x,
                              const float* __restrict__ scaleC,
                              const float* __restrict__ biasC,
                              float* __restrict__ out,
                              int Cdiv4) {
    int c4 = blockIdx.x * blockDim.x + threadIdx.x;
    if (c4 >= Cdiv4) return;
    long long i = (long long)blockIdx.y * (long long)Cdiv4 + (long long)c4;

    const v4f* x4 = (const v4f*)x;
    const v4f* s4 = (const v4f*)scaleC;
    const v4f* b4 = (const v4f*)biasC;

    v4f v = x4[i];
    v4f s = s4[c4];
    v4f b = b4[c4];
    v4f o;
    o.x = fmaf(v.x, s.x, b.x);
    o.y = fmaf(v.y, s.y, b.y);
    o.z = fmaf(v.z, s.z, b.z);
    o.w = fmaf(v.w, s.w, b.w);
    __builtin_nontemporal_store(o, ((v4f*)out) + i);
}

// ---------------------------------------------------------------------------
extern "C" void kernel_launch(void* const* d_in, const int* in_sizes, int n_in,
                              void* d_out, int out_size, void* d_ws, size_t ws_size,
                              hipStream_t stream) {
    const float* x     = (const float*)d_in[0];
    const int*   cg    = (const int*)d_in[1];
    const float* gamma = (const float*)d_in[2];
    const float* beta  = (const float*)d_in[3];
    float*       out   = (float*)d_out;

    const int C = in_sizes[1];
    const int G = in_sizes[2];
    const int B = in_sizes[0] / C;

    // Workspace layout (floats): s1[G] s2[G] cnt[G] scaleG[G] biasG[G] scaleC[C] biasC[C]
    float* w      = (float*)d_ws;
    float* s1     = w;
    float* s2     = w + G;
    float* cnt    = w + 2 * G;
    float* scaleG = w + 3 * G;
    float* biasG  = w + 4 * G;
    float* scaleC = w + 5 * G;
    float* biasC  = w + 5 * G + C;

    // 1) zero accumulators (3*G floats)
    {
        int n = 3 * G;
        gbn_zero<<<(n + 255) / 256, 256, 0, stream>>>(s1, n);
    }

    // 2) WMMA batch reduction + segment atomics: one wave per 16-channel tile,
    //    batch split 4 ways (rowsPerSlice kept a multiple of 4).
    {
        const int BSPLIT = 4;
        int rowsPerSlice = (B + BSPLIT - 1) / BSPLIT;
        rowsPerSlice = (rowsPerSlice + 3) & ~3;
        dim3 grid(C / 16, BSPLIT);
        gbn_colsum_wmma<<<grid, 32, 0, stream>>>(x, cg, s1, s2, cnt, B, C,
                                                 rowsPerSlice);
    }

    // 3) group params
    gbn_group_params<<<(G + 255) / 256, 256, 0, stream>>>(s1, s2, cnt, gamma, beta,
                                                          scaleG, biasG, G);

    // 3b) per-channel params
    gbn_channel_params<<<(C + 255) / 256, 256, 0, stream>>>(cg, scaleG, biasG,
                                                            scaleC, biasC, C);

    // 4) normalize (float4, 2D grid, NT stores)
    {
        int Cdiv4 = C / 4;
        dim3 grid((Cdiv4 + 255) / 256, B);
        gbn_normalize<<<grid, 256, 0, stream>>>(x, scaleC, biasC, out, Cdiv4);
    }
}